// SimpleStudent_72791105732705
// MI455X (gfx1250) — compile-verified
//
#include <hip/hip_runtime.h>
#include <hip/hip_bf16.h>
#include <math.h>

#define B_ROWS 2048
#define S_LEN  4096
#define D_DIM  20
#define H_DIM  10
#define NVOCAB 6
#define K_SEL  614

typedef __attribute__((ext_vector_type(2))) float v2f;
typedef __attribute__((ext_vector_type(8))) float v8f;

// ---------------------------------------------------------------------------
// Kernel 1: selector MLP on the 6 embedding-table rows -> scores; sort vocab
// ids by descending score (selection sort with swaps: not stable on ties).
// ---------------------------------------------------------------------------
__global__ void score_perm_kernel(const float* __restrict__ emb,
                                  const float* __restrict__ w1,
                                  const float* __restrict__ b1,
                                  const float* __restrict__ w2,
                                  const float* __restrict__ b2,
                                  int* __restrict__ perm)
{
  __shared__ float sc[NVOCAB];
  int v = threadIdx.x;
  if (v < NVOCAB) {
    float acc = 0.f;
    for (int j = 0; j < H_DIM; ++j) {
      float h = b1[j];
      for (int d = 0; d < D_DIM; ++d)
        h = fmaf(emb[v * D_DIM + d], w1[d * H_DIM + j], h);
      h = fmaxf(h, 0.f);
      acc = fmaf(h, w2[j], acc);
    }
    acc += b2[0];
    sc[v] = 1.f / (1.f + expf(-acc));
  }
  __syncthreads();
  if (threadIdx.x == 0) {
    int ord[NVOCAB] = {0, 1, 2, 3, 4, 5};
    for (int i = 0; i < NVOCAB; ++i) {
      int best = i;
      for (int j = i + 1; j < NVOCAB; ++j)
        if (sc[ord[j]] > sc[ord[best]]) best = j;
      int t = ord[i]; ord[i] = ord[best]; ord[best] = t;
    }
    for (int i = 0; i < NVOCAB; ++i) perm[i] = ord[i];
  }
}

// ---------------------------------------------------------------------------
// Kernel 2: one block per batch row. Histogram the 4096 tokens over 6 vocab
// values, compute the budgeted top-K selection per vocab rank, emit selected
// token positions (stable ascending order within each vocab), and produce the
// pooled vector  pooled[b] = sum_v selc[v]*emb[v] / K.
// ---------------------------------------------------------------------------
__global__ void __launch_bounds__(256)
topk_pool_kernel(const int* __restrict__ x,
                 const float* __restrict__ emb,
                 const int* __restrict__ perm,
                 float* __restrict__ pooled,
                 int* __restrict__ idx_out)
{
  const int b = blockIdx.x;
  const int t = threadIdx.x;            // 0..255, 16 contiguous tokens each
  __shared__ int cnt[NVOCAB][256];
  __shared__ int total[NVOCAB], selc[NVOCAB], basec[NVOCAB];

  const int* row = x + (size_t)b * S_LEN;
  int vals[16];
  const int4* row4 = (const int4*)(row + t * 16);
  #pragma unroll
  for (int q = 0; q < 4; ++q) {
    int4 p = row4[q];
    vals[q * 4 + 0] = p.x; vals[q * 4 + 1] = p.y;
    vals[q * 4 + 2] = p.z; vals[q * 4 + 3] = p.w;
  }

  // register histogram (compile-time vocab -> no scratch)
  int c0 = 0, c1 = 0, c2 = 0, c3 = 0, c4 = 0, c5 = 0;
  #pragma unroll
  for (int i = 0; i < 16; ++i) {
    int v = vals[i];
    c0 += (v == 0); c1 += (v == 1); c2 += (v == 2);
    c3 += (v == 3); c4 += (v == 4); c5 += (v == 5);
  }
  cnt[0][t] = c0; cnt[1][t] = c1; cnt[2][t] = c2;
  cnt[3][t] = c3; cnt[4][t] = c4; cnt[5][t] = c5;
  __syncthreads();

  // exclusive prefix over threads, per vocab (6 serial scans in parallel)
  if (t < NVOCAB) {
    int run = 0;
    for (int j = 0; j < 256; ++j) { int tmp = cnt[t][j]; cnt[t][j] = run; run += tmp; }
    total[t] = run;
  }
  __syncthreads();

  // budgeted take in score-rank order
  if (t == 0) {
    int budget = K_SEL, pos = 0;
    for (int r = 0; r < NVOCAB; ++r) {
      int v = perm[r];
      int take = min(total[v], budget);
      selc[v] = take; basec[v] = pos;
      pos += take; budget -= take;
    }
  }
  __syncthreads();

  // stable compaction: emit ascending positions per vocab
  int* out = idx_out + (size_t)b * K_SEL;
  #pragma unroll
  for (int v = 0; v < NVOCAB; ++v) {
    int ord = cnt[v][t];
    const int sc_ = selc[v];
    const int ba  = basec[v];
    #pragma unroll
    for (int i = 0; i < 16; ++i) {
      if (vals[i] == v) {
        if (ord < sc_) out[ba + ord] = t * 16 + i;
        ++ord;
      }
    }
  }

  // pooled mean via counts (emb row 5 is the zero padding row already)
  if (t < D_DIM) {
    float acc = 0.f;
    #pragma unroll
    for (int v = 0; v < NVOCAB; ++v)
      acc = fmaf((float)selc[v], emb[v * D_DIM + t], acc);
    pooled[(size_t)b * D_DIM + t] = acc * (1.0f / (float)K_SEL);
  }
}

// ---------------------------------------------------------------------------
// Kernel 3: processor MLP via V_WMMA_F32_16X16X4_F32.
// One wave (32 lanes) handles 16 batch rows. Layer 1: [16x20] @ [20x20] as
// 5 K-steps of 16x16x4 WMMA over two N-tiles (cols 0-15, 16-19 zero-padded).
// Layer 2: per-lane dot with proc_w2 + shfl_xor butterfly over each 16-lane
// half (C layout: rows 0-7 in lanes 0-15, rows 8-15 in lanes 16-31).
// ---------------------------------------------------------------------------
__global__ void __launch_bounds__(32)
proc_mlp_kernel(const float* __restrict__ pooled,
                const float* __restrict__ w1,
                const float* __restrict__ b1,
                const float* __restrict__ w2,
                const float* __restrict__ b2,
                float* __restrict__ pred)
{
  const int lane = threadIdx.x;
  const int m  = lane & 15;
  const int hi = lane >> 4;            // 0: K pair {0,1}; 1: K pair {2,3}
  const int rowBase = blockIdx.x * 16;
  const int n1 = 16 + m;
  const bool n1ok = (n1 < D_DIM);

  v8f acc0 = {}; v8f acc1 = {};
  #pragma unroll
  for (int kk = 0; kk < 5; ++kk) {
    const int k0 = kk * 4 + hi * 2;
    // A fragment: 16x4 tile of pooled rows [rowBase..rowBase+15], K=k0..k0+1
    const float* prow = pooled + (size_t)(rowBase + m) * D_DIM + k0;
    v2f a;  a.x = prow[0];  a.y = prow[1];
    // B fragments: 4x16 slices of proc_w1 (row-major [K=20][N=20])
    v2f bA; bA.x = w1[k0 * D_DIM + m];       bA.y = w1[(k0 + 1) * D_DIM + m];
    v2f bB; bB.x = n1ok ? w1[k0 * D_DIM + n1] : 0.f;
            bB.y = n1ok ? w1[(k0 + 1) * D_DIM + n1] : 0.f;
    acc0 = __builtin_amdgcn_wmma_f32_16x16x4_f32(false, a, false, bA, (short)0, acc0, false, false);
    acc1 = __builtin_amdgcn_wmma_f32_16x16x4_f32(false, a, false, bB, (short)0, acc1, false, false);
  }

  // bias + ReLU + dot with proc_w2 (lane owns columns m and 16+m)
  const float bias0 = b1[m];
  const float bias1 = n1ok ? b1[n1] : 0.f;
  const float w2a   = w2[m];
  const float w2b   = n1ok ? w2[n1] : 0.f;
  float part[8];
  #pragma unroll
  for (int j = 0; j < 8; ++j) {
    float p0 = fmaxf(acc0[j] + bias0, 0.f);
    float p1 = fmaxf(acc1[j] + bias1, 0.f);
    part[j] = p0 * w2a + p1 * w2b;
  }
  // butterfly reduce within each 16-lane half (masks < 16 keep halves apart)
  #pragma unroll
  for (int j = 0; j < 8; ++j) {
    float s = part[j];
    s += __shfl_xor(s, 1, 32);
    s += __shfl_xor(s, 2, 32);
    s += __shfl_xor(s, 4, 32);
    s += __shfl_xor(s, 8, 32);
    part[j] = s;
  }
  if (m == 0) {
    const float bb = b2[0];
    #pragma unroll
    for (int j = 0; j < 8; ++j) {
      float tl = part[j] + bb;
      pred[rowBase + hi * 8 + j] = 1.f / (1.f + expf(-tl));
    }
  }
}

// ---------------------------------------------------------------------------
extern "C" void kernel_launch(void* const* d_in, const int* in_sizes, int n_in,
                              void* d_out, int out_size, void* d_ws, size_t ws_size,
                              hipStream_t stream)
{
  const int*   x       = (const int*)  d_in[0];
  const float* emb     = (const float*)d_in[1];
  const float* sel_w1  = (const float*)d_in[2];
  const float* sel_b1  = (const float*)d_in[3];
  const float* sel_w2  = (const float*)d_in[4];
  const float* sel_b2  = (const float*)d_in[5];
  const float* proc_w1 = (const float*)d_in[6];
  const float* proc_b1 = (const float*)d_in[7];
  const float* proc_w2 = (const float*)d_in[8];
  const float* proc_b2 = (const float*)d_in[9];

  // outputs: prediction [2048] f32, then indices [2048*614] i32
  float* pred    = (float*)d_out;
  int*   idx_out = (int*)(pred + B_ROWS);

  // workspace: perm (6 ints) @0, pooled [2048*20] f32 @256B
  int*   perm   = (int*)d_ws;
  float* pooled = (float*)((char*)d_ws + 256);

  score_perm_kernel<<<1, 32, 0, stream>>>(emb, sel_w1, sel_b1, sel_w2, sel_b2, perm);
  topk_pool_kernel<<<B_ROWS, 256, 0, stream>>>(x, emb, perm, pooled, idx_out);
  proc_mlp_kernel<<<B_ROWS / 16, 32, 0, stream>>>(pooled, proc_w1, proc_b1,
                                                  proc_w2, proc_b2, pred);
}